// QFM_40759239639136
// MI455X (gfx1250) — compile-verified
//
#include <hip/hip_runtime.h>

// Problem constants (match reference)
constexpr int Bn = 16384;   // samples
constexpr int Fn = 39;      // features
constexpr int Vn = 100000;  // vocab per feature
constexpr int Dn = 128;     // embedding dim
constexpr int Kn = 256;     // codebook entries per feature
constexpr int Mn = 8;       // sub-codebooks (Q = D/M = 16 floats each)

typedef float v2f __attribute__((ext_vector_type(2)));
typedef float v8f __attribute__((ext_vector_type(8)));

// One wave32 per sample. Lane l owns dims 4l..4l+3 (all inside sub-codebook m = l/4).
__global__ __launch_bounds__(256) void qfm_fm_kernel(
    const int*   __restrict__ x,          // (B, F)
    const float* __restrict__ codebooks,  // (F*K, D)
    const int*   __restrict__ cb_index,   // (F*V, M)
    const float* __restrict__ lw,         // (F*V,)
    const float* __restrict__ bias,       // (1,)
    float*       __restrict__ out)        // (B,)
{
    const int lane = threadIdx.x & 31;
    const int wave = threadIdx.x >> 5;
    const int b    = blockIdx.x * 8 + wave;

    // ---- coalesced load of this sample's x row into lanes ----
    const long xrow = (long)b * Fn;
    const int  xa   = x[xrow + lane];                 // features 0..31
    const int  f2   = lane + 32;
    const int  f2c  = (f2 < Fn) ? f2 : (Fn - 1);      // clamp keeps reads in-bounds
    const int  xb   = x[xrow + f2c];                  // features 32..38 (lanes 0..6)

    // ---- linear term: lane l gathers feature l and feature l+32 ----
    float linp = __builtin_nontemporal_load(&lw[(long)xa + (long)lane * Vn]);
    {
        float l2 = __builtin_nontemporal_load(&lw[(long)xb + (long)f2c * Vn]);
        linp += (f2 < Fn) ? l2 : 0.0f;                // select, no divergence
    }

    const int    mlane  = lane >> 2;                  // sub-codebook for my dims
    const float* cbbase = codebooks + (lane << 2);    // my 16B slot within a row

    float4 s = make_float4(0.f, 0.f, 0.f, 0.f);       // Σ_f e[d]   (4 dims)
    float  q = 0.f;                                   // Σ_f e[d]^2 (4 dims)

    // Fully unrolled: 39 independent cb_index gathers can all be in flight.
#pragma unroll
    for (int f = 0; f < Fn; ++f) {
        const int xv = (f < 32) ? __builtin_amdgcn_readlane(xa, f)
                                : __builtin_amdgcn_readlane(xb, f - 32);
        const long xo  = (long)xv + (long)f * Vn;                     // row in cb_index
        const int  cbm = __builtin_nontemporal_load(&cb_index[xo * Mn + mlane]);
        const long row = (long)(cbm + f * Kn) * Dn;
        const float4 e = *reinterpret_cast<const float4*>(cbbase + row);
        s.x += e.x; s.y += e.y; s.z += e.z; s.w += e.w;
        q = fmaf(e.x, e.x, q); q = fmaf(e.y, e.y, q);
        q = fmaf(e.z, e.z, q); q = fmaf(e.w, e.w, q);
    }

    // per-lane FM partial + linear partial
    const float p = s.x * s.x + s.y * s.y + s.z * s.z + s.w * s.w;
    float r = 0.5f * (p - q) + linp;

    // ---- wave-wide sum of r on the matrix pipe (exact, layout-robust) ----
    // Level 1: B = ones  =>  D[i][*] = rowsum(A_i).  A row i lives in lanes
    // {i, 16+i} across both A VGPRs (ISA 7.12.2), so A=(r,r) gives
    // v_i = 2*(r_i + r_{16+i}) replicated across row i of D.
    v2f av;  av[0] = r;    av[1] = r;
    v2f onev; onev[0] = 1.0f; onev[1] = 1.0f;
    v8f zc = {};
    v8f d1 = __builtin_amdgcn_wmma_f32_16x16x4_f32(
        false, av, false, onev, (short)0, zc, false, false);

    // Level 2: C/D VGPR t holds rows {t, 8+t} (N = lane%16).  Feeding VGPR
    // pairs (2t, 2t+1) back as A and B=ones accumulates v_{2t}+v_{2t+1}+
    // v_{8+2t}+v_{9+2t} into every element; chaining C over t=0..3 sums all
    // sixteen v_i:  acc[*] = Σ v_i = 2 * Σ_lane r   (uniform in every lane).
    v8f acc = {};
#pragma unroll
    for (int t = 0; t < 4; ++t) {
        v2f a2; a2[0] = d1[2 * t]; a2[1] = d1[2 * t + 1];
        acc = __builtin_amdgcn_wmma_f32_16x16x4_f32(
            false, a2, false, onev, (short)0, acc, false, false);
    }
    const float wave_sum = 0.5f * acc[0];

    if (lane == 0) out[b] = wave_sum + bias[0];
}

extern "C" void kernel_launch(void* const* d_in, const int* in_sizes, int n_in,
                              void* d_out, int out_size, void* d_ws, size_t ws_size,
                              hipStream_t stream) {
    (void)in_sizes; (void)n_in; (void)d_ws; (void)ws_size; (void)out_size;
    const int*   x         = (const int*)  d_in[0];
    const float* codebooks = (const float*)d_in[1];
    const int*   cb_index  = (const int*)  d_in[2];
    const float* lw        = (const float*)d_in[3];
    const float* bias      = (const float*)d_in[4];
    float*       out       = (float*)d_out;

    dim3 block(256);            // 8 wave32 waves = 8 samples per block
    dim3 grid(Bn / 8);          // 2048 blocks
    qfm_fm_kernel<<<grid, block, 0, stream>>>(x, codebooks, cb_index, lw, bias, out);
}